// GaussianRasterizer_223338300187
// MI455X (gfx1250) — compile-verified
//
#include <hip/hip_runtime.h>
#include <cmath>

#ifndef __has_builtin
#define __has_builtin(x) 0
#endif

#define BLOCK 256
#define GPT 4  // gaussians per thread (4 g = 48 B = 3 x float4, 16B aligned)

#if __has_builtin(__builtin_amdgcn_global_load_async_to_lds_b128)
#define HAVE_ASYNC_LDS 1
#else
#define HAVE_ASYNC_LDS 0
#endif

#if HAVE_ASYNC_LDS
typedef int v4i __attribute__((vector_size(4 * sizeof(int))));
typedef __attribute__((address_space(1))) v4i gv4;   // global
typedef __attribute__((address_space(3))) v4i lv4;   // LDS
#endif

// ---------------------------------------------------------------- init winner
__global__ __launch_bounds__(BLOCK) void gs_init_winner(int* __restrict__ winner, int P) {
    int p = blockIdx.x * blockDim.x + threadIdx.x;
    if (p < P) winner[p] = -1;
}

// ---------------------------------------------------------------- scatter-max
__global__ __launch_bounds__(BLOCK) void gs_scatter_max(const float* __restrict__ pos,
                                                        int* __restrict__ winner,
                                                        int N, int S) {
#if HAVE_ASYNC_LDS
    __shared__ float stage[BLOCK * 12];
#endif
    int t = blockIdx.x * blockDim.x + threadIdx.x;
    long long g0 = (long long)t * GPT;
    if (g0 >= N) return;

    const float half = 0.5f * (float)S;
    int   cnt = (N - g0 >= GPT) ? GPT : (int)(N - g0);
    float xs[GPT], ys[GPT];

    if (cnt == GPT) {
#if HAVE_ASYNC_LDS
        // CDNA5 async global->LDS staging (ASYNCcnt), then LDS reads.
        char*  src  = (char*)(pos + (size_t)g0 * 3);
        float* slot = stage + (size_t)threadIdx.x * 12;
        asm volatile("" ::: "memory");  // keep issue order compiler-visible
        __builtin_amdgcn_global_load_async_to_lds_b128((gv4*)(src +  0), (lv4*)(slot + 0), 0, 0);
        __builtin_amdgcn_global_load_async_to_lds_b128((gv4*)(src + 16), (lv4*)(slot + 4), 0, 0);
        __builtin_amdgcn_global_load_async_to_lds_b128((gv4*)(src + 32), (lv4*)(slot + 8), 0, 0);
#if __has_builtin(__builtin_amdgcn_s_wait_asynccnt)
        __builtin_amdgcn_s_wait_asynccnt(0);
#else
        asm volatile("s_wait_asynccnt 0" ::: "memory");
#endif
        asm volatile("" ::: "memory");  // block LDS-load hoisting above the wait
        float a0 = slot[0], a1 = slot[1];
        float a3 = slot[3], b0 = slot[4];
        float b2 = slot[6], b3 = slot[7];
        float c1 = slot[9], c2 = slot[10];
        xs[0] = a0; ys[0] = a1;
        xs[1] = a3; ys[1] = b0;
        xs[2] = b2; ys[2] = b3;
        xs[3] = c1; ys[3] = c2;
#else
        // Coalesced B128 path: 3 x float4 covers 4 gaussians, 16B-aligned.
        const float4* p4 = (const float4*)(pos + (size_t)g0 * 3);
        float4 a = p4[0];  // x0 y0 z0 x1
        float4 b = p4[1];  // y1 z1 x2 y2
        float4 c = p4[2];  // z2 x3 y3 z3
        xs[0] = a.x; ys[0] = a.y;
        xs[1] = a.w; ys[1] = b.x;
        xs[2] = b.z; ys[2] = b.w;
        xs[3] = c.y; ys[3] = c.z;
#endif
    } else {
        for (int k = 0; k < cnt; ++k) {
            xs[k] = pos[(size_t)(g0 + k) * 3 + 0];
            ys[k] = pos[(size_t)(g0 + k) * 3 + 1];
        }
    }

#pragma unroll
    for (int k = 0; k < GPT; ++k) {
        if (k >= cnt) break;
        // trunc-toward-zero matches astype(int32); clip to [0, S-1]
        int px = (int)((xs[k] + 1.0f) * half);
        int py = (int)((ys[k] + 1.0f) * half);
        px = min(max(px, 0), S - 1);
        py = min(max(py, 0), S - 1);
        atomicMax(&winner[py * S + px], (int)(g0 + k));  // global_atomic_max_i32, L2-resident
    }
}

// ---------------------------------------------------------------- resolve/gather
__global__ __launch_bounds__(BLOCK) void gs_resolve(const int* __restrict__ winner,
                                                    const float* __restrict__ colors,
                                                    const float* __restrict__ opac,
                                                    float* __restrict__ out, int P) {
    int p = blockIdx.x * blockDim.x + threadIdx.x;
    if (p >= P) return;
    int w = __builtin_nontemporal_load(&winner[p]);  // single-use, don't cache
    float r = 0.f, g = 0.f, b = 0.f, a = 0.f;
    if (w >= 0) {
        float  o  = opac[w];
        size_t cb = (size_t)w * 3;
        r = colors[cb + 0] * o;
        g = colors[cb + 1] * o;
        b = colors[cb + 2] * o;
        a = o;
    }
    // planar (4,S,S) output; write-once -> non-temporal so L2 keeps gather tables
    __builtin_nontemporal_store(r, out + p);
    __builtin_nontemporal_store(g, out + (size_t)P + p);
    __builtin_nontemporal_store(b, out + 2 * (size_t)P + p);
    __builtin_nontemporal_store(a, out + 3 * (size_t)P + p);
}

// ---------------------------------------------------------------- launcher
extern "C" void kernel_launch(void* const* d_in, const int* in_sizes, int n_in,
                              void* d_out, int out_size, void* d_ws, size_t ws_size,
                              hipStream_t stream) {
    const float* positions = (const float*)d_in[0];
    const float* colors    = (const float*)d_in[1];
    const float* opac      = (const float*)d_in[2];
    // d_in[3] camera_params: unused by the reference math
    // d_in[4] image_size: recover S from out_size instead (device scalar)

    int N = in_sizes[0] / 3;
    int P = out_size / 4;          // S*S pixels
    int S = (int)sqrtf((float)P);
    while ((long long)(S + 1) * (S + 1) <= P) ++S;
    while ((long long)S * S > P) --S;

    // winner buffer: prefer workspace; fall back to the alpha plane of d_out
    // (resolve reads winner[p] before overwriting alpha[p] in the same thread,
    //  and the other three planes don't overlap it -> race-free).
    int* winner = (ws_size >= (size_t)P * sizeof(int))
                      ? (int*)d_ws
                      : (int*)((float*)d_out + 3 * (size_t)P);

    int gridP = (P + BLOCK - 1) / BLOCK;
    gs_init_winner<<<gridP, BLOCK, 0, stream>>>(winner, P);

    int nThreads = (N + GPT - 1) / GPT;
    int gridN    = (nThreads + BLOCK - 1) / BLOCK;
    gs_scatter_max<<<gridN, BLOCK, 0, stream>>>(positions, winner, N, S);

    gs_resolve<<<gridP, BLOCK, 0, stream>>>(winner, colors, opac, (float*)d_out, P);
}